// HyperQMDPLayer_81028853006779
// MI455X (gfx1250) — compile-verified
//
#include <hip/hip_runtime.h>
#include <hip/hip_bf16.h>
#include <math.h>

// dims
#define S_DIM 256
#define A_DIM 16
#define H_DIM 30
#define HY_DIM 64
#define B_DIM 64
#define T_DIM 64
#define KI_DIM (A_DIM * S_DIM)              /* 4096 (k,i) pairs          */
#define ASS (A_DIM * S_DIM * S_DIM)         /* 1,048,576 elems per batch */
#define EPSV 1e-6f

typedef __attribute__((ext_vector_type(2))) float v2f;
typedef __attribute__((ext_vector_type(8))) float v8f;

struct BF4 { __hip_bfloat162 lo, hi; };     // 8-byte bf16x4 load

// ---------------------------------------------------------------------------
// Phase 0: horizon distribution  h_dist[b,h]  (tiny; one wave-pair)
// rate = exp(clip(tau + z.tau_ow, -8, 8));  softmax_h(k*log(rate) - rate - lgamma(k+1))
// ---------------------------------------------------------------------------
__global__ void horizon_kernel(const float* __restrict__ z,
                               const float* __restrict__ tau,
                               const float* __restrict__ tau_ow,
                               float* __restrict__ h_dist) {
  int b = threadIdx.x;
  if (b >= B_DIM) return;
  float dot = 0.f;
  for (int hy = 0; hy < HY_DIM; ++hy) dot += z[b * HY_DIM + hy] * tau_ow[hy];
  float x = fminf(8.f, fmaxf(-8.f, tau[0] + dot));  // = log(rate) after clip
  float rate = __expf(x);
  float lp[H_DIM];
  float lf = 0.f, m = -1e30f;
  for (int k = 1; k <= H_DIM; ++k) {
    if (k > 1) lf += __logf((float)k);               // lgamma(k+1)
    float v = (float)k * x - rate - lf;
    lp[k - 1] = v;
    m = fmaxf(m, v);
  }
  float ssum = 0.f;
  for (int h = 0; h < H_DIM; ++h) { lp[h] = __expf(lp[h] - m); ssum += lp[h]; }
  float inv = 1.f / ssum;
  for (int h = 0; h < H_DIM; ++h) h_dist[b * H_DIM + h] = lp[h] * inv;
}

// ---------------------------------------------------------------------------
// Phase 1: transition = softmax_j( z @ Wow^T  + w )  stored bf16 (134 MB -> L2)
// GEMM M=64(b) K=64(hy) N=1M(k,i,j) via V_WMMA_F32_16X16X4_F32 chains.
// Block: 512 thr = 16 waves; block tile = 16 batches x 256 j (one (k,i) row
// group) so the softmax over j is local. Grid = (4096 ki, 4 btile).
// ---------------------------------------------------------------------------
__global__ __launch_bounds__(512)
void transition_kernel(const float* __restrict__ z,
                       const float* __restrict__ w,
                       const float* __restrict__ wow,
                       __hip_bfloat16* __restrict__ trans) {
  __shared__ float tile[16 * 256];  // [b-row][j]
  const int ki    = blockIdx.x;     // (k,i) flattened, 0..4095
  const int btile = blockIdx.y;     // 0..3
  const int tid   = threadIdx.x;
  const int wave  = tid >> 5;       // 0..15: j-subtile
  const int lane  = tid & 31;
  const int l     = lane & 15;
  const bool hiHalf = lane >= 16;

  const int bRow = btile * 16 + l;                 // A-matrix row (batch)
  const int nCol = ki * 256 + wave * 16 + l;       // B-matrix col (n=(k,i,j))

  v8f d = {0.f, 0.f, 0.f, 0.f, 0.f, 0.f, 0.f, 0.f};
  const int hyOff = hiHalf ? 2 : 0;
#pragma unroll
  for (int kk = 0; kk < 16; ++kk) {                // K = 64 = 16 x 4
    int hy = kk * 4 + hyOff;
    v2f a, bb;
    a.x  = z[bRow * HY_DIM + hy];
    a.y  = z[bRow * HY_DIM + hy + 1];
    bb.x = wow[(size_t)nCol * HY_DIM + hy];        // Wow^T fragment
    bb.y = wow[(size_t)nCol * HY_DIM + hy + 1];
    d = __builtin_amdgcn_wmma_f32_16x16x4_f32(false, a, false, bb, (short)0, d,
                                              false, false);
  }
  // add base weights w (broadcast over batch: depends only on column n)
  float wv = w[(size_t)ki * 256 + wave * 16 + l];
#pragma unroll
  for (int r = 0; r < 8; ++r) {
    int row = hiHalf ? (r + 8) : r;                // D layout: VGPR r -> M=r / r+8
    tile[row * 256 + wave * 16 + l] = d[r] + wv;
  }
  __syncthreads();

  // row-wise softmax over 256 j; wave r owns batch-row r
  const int row = wave;
  float vals[8];
  float m = -1e30f;
#pragma unroll
  for (int q = 0; q < 8; ++q) {
    vals[q] = tile[row * 256 + lane + 32 * q];
    m = fmaxf(m, vals[q]);
  }
#pragma unroll
  for (int off = 16; off > 0; off >>= 1) m = fmaxf(m, __shfl_xor(m, off, 32));
  float ssum = 0.f;
#pragma unroll
  for (int q = 0; q < 8; ++q) { vals[q] = __expf(vals[q] - m); ssum += vals[q]; }
#pragma unroll
  for (int off = 16; off > 0; off >>= 1) ssum += __shfl_xor(ssum, off, 32);
  float inv = 1.f / ssum;

  const int bOut = btile * 16 + row;
  __hip_bfloat16* dst = trans + (size_t)bOut * ASS + (size_t)ki * 256;
#pragma unroll
  for (int q = 0; q < 8; ++q)
    dst[lane + 32 * q] = __float2bfloat16(vals[q] * inv);
}

// ---------------------------------------------------------------------------
// Phase 2: persistent scan. One block per batch element (scan is independent
// per b). 512 threads, T-loop inside. Per step:
//   s_next[j] = sum_ki (exp(logp_u[t,b,k]) * b_t[i]) * trans[b,ki,j]   (bf16 GEMV)
//   b_next    = softmax(log(s_next+eps) + logp_o[t,b,:])
//   alpha_pi  = sum_h softmax_k(<b_next, value[h,b,k,:]>) * h_dist[b,h]
// ---------------------------------------------------------------------------
__global__ __launch_bounds__(512)
void scan_kernel(const float* __restrict__ logp_o,
                 const float* __restrict__ logp_u,
                 const float* __restrict__ value,
                 const float* __restrict__ b_in,
                 const __hip_bfloat16* __restrict__ trans,
                 const float* __restrict__ h_dist,
                 float* __restrict__ out_b,    // [T,B,S]
                 float* __restrict__ out_pi) { // [T,B,A]
  __shared__ float cvec[KI_DIM];         // 16 KB: a_t[k]*b_t[i]
  __shared__ float red[8 * 256];         //  8 KB: split-K partials
  __shared__ float bcur[S_DIM];
  __shared__ float lgts[H_DIM * A_DIM];  // 480 planning logits
  __shared__ float hloc[H_DIM];
  __shared__ float rtmp[16];

  const int b    = blockIdx.x;
  const int tid  = threadIdx.x;
  const int lane = tid & 31;
  const int wid  = tid >> 5;

  if (tid < S_DIM) bcur[tid] = b_in[b * S_DIM + tid];
  if (tid < H_DIM) hloc[tid] = h_dist[b * H_DIM + tid];
  __syncthreads();

  const __hip_bfloat16* tr = trans + (size_t)b * ASS;
  const int s  = tid >> 6;        // reduction slice 0..7
  const int jb = (tid & 63) * 4;  // 4 consecutive next-states

  for (int t = 0; t < T_DIM; ++t) {
    // combined weights (alpha_a = exp(log_softmax) = exp)
    for (int idx = tid; idx < KI_DIM; idx += 512) {
      int k = idx >> 8, i = idx & 255;
      cvec[idx] = __expf(logp_u[((size_t)t * B_DIM + b) * A_DIM + k]) * bcur[i];
    }
    __syncthreads();

    // bf16 GEMV: coalesced 8B loads, split-K by 8
    float a0 = 0.f, a1 = 0.f, a2 = 0.f, a3 = 0.f;
    for (int ki = s; ki < KI_DIM; ki += 8) {
      float cv = cvec[ki];
      BF4 v = *(const BF4*)(tr + (size_t)ki * 256 + jb);
      float2 f0 = __bfloat1622float2(v.lo);
      float2 f1 = __bfloat1622float2(v.hi);
      a0 = fmaf(cv, f0.x, a0);
      a1 = fmaf(cv, f0.y, a1);
      a2 = fmaf(cv, f1.x, a2);
      a3 = fmaf(cv, f1.y, a3);
    }
    red[s * 256 + jb + 0] = a0;
    red[s * 256 + jb + 1] = a1;
    red[s * 256 + jb + 2] = a2;
    red[s * 256 + jb + 3] = a3;
    __syncthreads();

    // belief softmax over 256 states (waves 0..7 active)
    float val = 0.f, e = 0.f;
    if (tid < S_DIM) {
      float sn = 0.f;
#pragma unroll
      for (int q = 0; q < 8; ++q) sn += red[q * 256 + tid];
      val = __logf(sn + EPSV) + logp_o[((size_t)t * B_DIM + b) * S_DIM + tid];
      float m = val;
#pragma unroll
      for (int off = 16; off > 0; off >>= 1) m = fmaxf(m, __shfl_xor(m, off, 32));
      if (lane == 0) rtmp[wid] = m;
    }
    __syncthreads();
    float M = fmaxf(fmaxf(fmaxf(rtmp[0], rtmp[1]), fmaxf(rtmp[2], rtmp[3])),
                    fmaxf(fmaxf(rtmp[4], rtmp[5]), fmaxf(rtmp[6], rtmp[7])));
    if (tid < S_DIM) {
      e = __expf(val - M);
      float ss = e;
#pragma unroll
      for (int off = 16; off > 0; off >>= 1) ss += __shfl_xor(ss, off, 32);
      if (lane == 0) rtmp[8 + wid] = ss;
    }
    __syncthreads();
    float SUM = ((rtmp[8] + rtmp[9]) + (rtmp[10] + rtmp[11])) +
                ((rtmp[12] + rtmp[13]) + (rtmp[14] + rtmp[15]));
    if (tid < S_DIM) {
      float bn = e / SUM;
      bcur[tid] = bn;
      out_b[((size_t)t * B_DIM + b) * S_DIM + tid] = bn;
    }
    __syncthreads();

    // planning: 480 dot products of length 256 (value is L2 resident)
    for (int p = tid; p < H_DIM * A_DIM; p += 512) {
      int h = p >> 4, k = p & 15;
      const float4* vp = (const float4*)(value +
          (((size_t)h * B_DIM + b) * A_DIM + k) * S_DIM);
      float dsum = 0.f;
#pragma unroll 4
      for (int i4 = 0; i4 < 64; ++i4) {
        float4 vv = vp[i4];
        dsum += vv.x * bcur[i4 * 4 + 0] + vv.y * bcur[i4 * 4 + 1] +
                vv.z * bcur[i4 * 4 + 2] + vv.w * bcur[i4 * 4 + 3];
      }
      lgts[p] = dsum;
    }
    __syncthreads();
    if (tid < H_DIM) {  // softmax over actions, scale by horizon weight
      float m = -1e30f;
#pragma unroll
      for (int k = 0; k < A_DIM; ++k) m = fmaxf(m, lgts[tid * 16 + k]);
      float ss = 0.f;
      float ev[A_DIM];
#pragma unroll
      for (int k = 0; k < A_DIM; ++k) {
        ev[k] = __expf(lgts[tid * 16 + k] - m);
        ss += ev[k];
      }
      float sc = hloc[tid] / ss;
#pragma unroll
      for (int k = 0; k < A_DIM; ++k) lgts[tid * 16 + k] = ev[k] * sc;
    }
    __syncthreads();
    if (tid < A_DIM) {  // mix over horizons
      float api = 0.f;
#pragma unroll
      for (int h = 0; h < H_DIM; ++h) api += lgts[h * 16 + tid];
      out_pi[((size_t)t * B_DIM + b) * A_DIM + tid] = api;
    }
    __syncthreads();
  }
}

// ---------------------------------------------------------------------------
extern "C" void kernel_launch(void* const* d_in, const int* in_sizes, int n_in,
                              void* d_out, int out_size, void* d_ws,
                              size_t ws_size, hipStream_t stream) {
  const float* logp_o = (const float*)d_in[0];
  const float* logp_u = (const float*)d_in[1];
  const float* value  = (const float*)d_in[2];
  const float* z      = (const float*)d_in[3];
  const float* b_in   = (const float*)d_in[4];
  const float* w      = (const float*)d_in[5];
  const float* wow    = (const float*)d_in[6];
  const float* tau    = (const float*)d_in[7];
  const float* tau_ow = (const float*)d_in[8];

  // workspace: bf16 transition (134 MB, fits MI455X 192 MB L2) + h_dist
  __hip_bfloat16* trans = (__hip_bfloat16*)d_ws;
  float* h_dist = (float*)((char*)d_ws +
                           (size_t)B_DIM * ASS * sizeof(__hip_bfloat16));

  float* out_b  = (float*)d_out;
  float* out_pi = out_b + (size_t)T_DIM * B_DIM * S_DIM;

  hipLaunchKernelGGL(horizon_kernel, dim3(1), dim3(64), 0, stream,
                     z, tau, tau_ow, h_dist);
  hipLaunchKernelGGL(transition_kernel, dim3(KI_DIM, 4), dim3(512), 0, stream,
                     z, w, wow, trans);
  hipLaunchKernelGGL(scan_kernel, dim3(B_DIM), dim3(512), 0, stream,
                     logp_o, logp_u, value, b_in, trans, h_dist, out_b, out_pi);
}